// Node_GCN_33397665693788
// MI455X (gfx1250) — compile-verified
//
#include <hip/hip_runtime.h>
#include <stdint.h>

typedef float v2f __attribute__((ext_vector_type(2)));
typedef float v8f __attribute__((ext_vector_type(8)));

#define NN   100000
#define EE   600000
#define F_IN 128
#define HID  256
#define F_OUT 64

// ---------------- elementwise / prep kernels ----------------

__global__ void fill_kernel(float* __restrict__ p, float v, long long n) {
    long long i = (long long)blockIdx.x * blockDim.x + threadIdx.x;
    if (i < n) p[i] = v;
}

__global__ void deg_scatter_kernel(const long long* __restrict__ col,
                                   const float* __restrict__ ew,
                                   float* __restrict__ deg, int e) {
    int i = blockIdx.x * blockDim.x + threadIdx.x;
    if (i < e) atomicAdd(&deg[(int)col[i]], ew[i]);
}

// deg -> dinv (in place), selfnorm = dinv^2
__global__ void dinv_kernel(float* __restrict__ deg,
                            float* __restrict__ selfnorm, int n) {
    int i = blockIdx.x * blockDim.x + threadIdx.x;
    if (i >= n) return;
    float d = deg[i];
    float di = d > 0.0f ? rsqrtf(d) : 0.0f;
    deg[i] = di;
    selfnorm[i] = di * di;
}

__global__ void norm_kernel(const long long* __restrict__ row,
                            const long long* __restrict__ col,
                            const float* __restrict__ ew,
                            const float* __restrict__ dinv,
                            float* __restrict__ norm, int e) {
    int i = blockIdx.x * blockDim.x + threadIdx.x;
    if (i >= e) return;
    norm[i] = dinv[(int)row[i]] * ew[i] * dinv[(int)col[i]];
}

// ---------------- fp32 WMMA GEMM: C = A(MxK) @ B(KxNOUT) [+ bias] ----------------
// One wave computes a 16 x (16*NTILES) strip using V_WMMA_F32_16X16X4_F32.
// A frag (16x4 f32): lanes 0-15 -> K=k,k+1 ; lanes 16-31 -> K=k+2,k+3 (M = lane&15).
// B frag (4x16 f32): same K-half split; N striped over lane&15.
// C/D  (16x16 f32): VGPR r holds M = r + 8*(lane>>4), N = lane&15.
// Explicit double-buffering: issue next k-step's load batch (1xA b64 + NTILES B pairs)
// BEFORE the current WMMA block so the matrix pipe overlaps load latency.
template <int NTILES>
__global__ __launch_bounds__(256) void gemm_wmma_f32(
    const float* __restrict__ A, const float* __restrict__ B,
    const float* __restrict__ bias, float* __restrict__ C,
    int M, int K, int NOUT) {
    const int wave    = (blockIdx.x * blockDim.x + threadIdx.x) >> 5;
    const int nblocks = NOUT / (NTILES * 16);
    const int total   = (M / 16) * nblocks;
    if (wave >= total) return;

    const int mtile = wave / nblocks;
    const int nblk  = wave % nblocks;
    const int lane  = threadIdx.x & 31;
    const int half  = lane >> 4;   // 0: K lo pair, 1: K hi pair
    const int l15   = lane & 15;

    const float* __restrict__ arow  = A + (size_t)(mtile * 16 + l15) * K + half * 2;
    const float* __restrict__ bbase = B + (size_t)(half * 2) * NOUT + nblk * NTILES * 16 + l15;

    v8f acc[NTILES];
    const v8f vzero = {};
#pragma unroll
    for (int t = 0; t < NTILES; ++t) acc[t] = vzero;

    // ---- prologue: load k = 0 fragments ----
    v2f a_cur = *(const v2f*)(arow);
    v2f b_cur[NTILES];
#pragma unroll
    for (int t = 0; t < NTILES; ++t) {
        b_cur[t].x = bbase[t * 16];
        b_cur[t].y = bbase[t * 16 + NOUT];
    }

    // ---- steady state: prefetch k+4 batch, then NTILES back-to-back WMMAs on k ----
    for (int k = 0; k < K - 4; k += 4) {
        const int kn = k + 4;
        v2f a_nxt = *(const v2f*)(arow + kn);
        const float* __restrict__ bp = bbase + (size_t)kn * NOUT;
        v2f b_nxt[NTILES];
#pragma unroll
        for (int t = 0; t < NTILES; ++t) {
            b_nxt[t].x = bp[t * 16];
            b_nxt[t].y = bp[t * 16 + NOUT];
        }
#pragma unroll
        for (int t = 0; t < NTILES; ++t) {
            acc[t] = __builtin_amdgcn_wmma_f32_16x16x4_f32(
                false, a_cur, false, b_cur[t], (short)0, acc[t], false, false);
        }
        a_cur = a_nxt;
#pragma unroll
        for (int t = 0; t < NTILES; ++t) b_cur[t] = b_nxt[t];
    }

    // ---- epilogue: last k-step ----
#pragma unroll
    for (int t = 0; t < NTILES; ++t) {
        acc[t] = __builtin_amdgcn_wmma_f32_16x16x4_f32(
            false, a_cur, false, b_cur[t], (short)0, acc[t], false, false);
    }

    const size_t base = (size_t)(mtile * 16 + half * 8) * NOUT + nblk * NTILES * 16 + l15;
#pragma unroll
    for (int t = 0; t < NTILES; ++t) {
        const float bv = bias ? bias[nblk * NTILES * 16 + t * 16 + l15] : 0.0f;
#pragma unroll
        for (int r = 0; r < 8; ++r) {
            C[base + (size_t)r * NOUT + t * 16] = acc[t][r] + bv;
        }
    }
}

// ---------------- edge aggregation: agg[col] += h[row] * norm ----------------
// One wave per edge; lanes stride the 256 features (8 coalesced chunks).
// h and agg (102 MB each) are L2-resident (192 MB), so gathers + f32 atomics hit L2.
__global__ __launch_bounds__(256) void scatter_kernel(
    const float* __restrict__ h,
    const long long* __restrict__ row, const long long* __restrict__ col,
    const float* __restrict__ norm, float* __restrict__ agg, int e, int F) {
    const int w = (blockIdx.x * blockDim.x + threadIdx.x) >> 5;
    if (w >= e) return;
    const int lane = threadIdx.x & 31;
    const int r = (int)row[w];
    const int c = (int)col[w];
    const float nrm = norm[w];
    const float* __restrict__ src = h + (size_t)r * F;
    float* __restrict__ dst = agg + (size_t)c * F;
    for (int f = lane; f < F; f += 32)
        atomicAdd(&dst[f], src[f] * nrm);
}

// agg = relu(agg + h*selfnorm + bias)   (self-loop folded in, avoids N*F atomics)
__global__ void epilogue_kernel(float* __restrict__ agg,
                                const float* __restrict__ h,
                                const float* __restrict__ selfnorm,
                                const float* __restrict__ bias,
                                int n, int F) {
    long long i = (long long)blockIdx.x * blockDim.x + threadIdx.x;
    if (i >= (long long)n * F) return;
    int node = (int)(i >> 8);      // F == 256
    int f    = (int)(i & 255);
    float v = agg[i] + h[i] * selfnorm[node] + bias[f];
    agg[i] = v > 0.0f ? v : 0.0f;
}

// ---------------- launcher ----------------

extern "C" void kernel_launch(void* const* d_in, const int* in_sizes, int n_in,
                              void* d_out, int out_size, void* d_ws, size_t ws_size,
                              hipStream_t stream) {
    const float*      x   = (const float*)d_in[0];
    const long long*  ei  = (const long long*)d_in[1];  // [0..E): row, [E..2E): col
    const float*      ew  = (const float*)d_in[2];
    const float*      W1  = (const float*)d_in[3];
    const float*      b1  = (const float*)d_in[4];
    const float*      W2  = (const float*)d_in[5];
    const float*      b2  = (const float*)d_in[6];
    const float*      Wfc = (const float*)d_in[7];
    const float*      bfc = (const float*)d_in[8];
    float*            out = (float*)d_out;

    const long long* row = ei;
    const long long* col = ei + EE;

    float* dinv     = (float*)d_ws;                 // N
    float* selfnorm = dinv + NN;                    // N
    float* norm     = selfnorm + NN;                // E
    float* bufA     = norm + EE;                    // N*HID
    float* bufB     = bufA + (size_t)NN * HID;      // N*HID

    const int B256 = 256;
    const long long nh = (long long)NN * HID;

    // --- normalization factors (deg lives in dinv buffer, self-loop weight = 1) ---
    fill_kernel<<<(NN + B256 - 1) / B256, B256, 0, stream>>>(dinv, 1.0f, NN);
    deg_scatter_kernel<<<(EE + B256 - 1) / B256, B256, 0, stream>>>(col, ew, dinv, EE);
    dinv_kernel<<<(NN + B256 - 1) / B256, B256, 0, stream>>>(dinv, selfnorm, NN);
    norm_kernel<<<(EE + B256 - 1) / B256, B256, 0, stream>>>(row, col, ew, dinv, norm, EE);

    // --- layer 1: h = x @ W1 ; agg ; relu(+b1) ---
    {
        int waves = (NN / 16) * (HID / (8 * 16));
        int blocks = (waves * 32 + B256 - 1) / B256;
        gemm_wmma_f32<8><<<blocks, B256, 0, stream>>>(x, W1, nullptr, bufA, NN, F_IN, HID);
    }
    fill_kernel<<<(int)((nh + B256 - 1) / B256), B256, 0, stream>>>(bufB, 0.0f, nh);
    scatter_kernel<<<(EE * 32 + B256 - 1) / B256, B256, 0, stream>>>(bufA, row, col, norm, bufB, EE, HID);
    epilogue_kernel<<<(int)((nh + B256 - 1) / B256), B256, 0, stream>>>(bufB, bufA, selfnorm, b1, NN, HID);

    // --- layer 2: h = act1 @ W2 ; agg ; relu(+b2) ---
    {
        int waves = (NN / 16) * (HID / (8 * 16));
        int blocks = (waves * 32 + B256 - 1) / B256;
        gemm_wmma_f32<8><<<blocks, B256, 0, stream>>>(bufB, W2, nullptr, bufA, NN, HID, HID);
    }
    fill_kernel<<<(int)((nh + B256 - 1) / B256), B256, 0, stream>>>(bufB, 0.0f, nh);
    scatter_kernel<<<(EE * 32 + B256 - 1) / B256, B256, 0, stream>>>(bufA, row, col, norm, bufB, EE, HID);
    epilogue_kernel<<<(int)((nh + B256 - 1) / B256), B256, 0, stream>>>(bufB, bufA, selfnorm, b2, NN, HID);

    // --- FC: out = act2 @ Wfc + bfc ---
    {
        int waves = (NN / 16) * (F_OUT / (4 * 16));
        int blocks = (waves * 32 + B256 - 1) / B256;
        gemm_wmma_f32<4><<<blocks, B256, 0, stream>>>(bufB, Wfc, bfc, out, NN, HID, F_OUT);
    }
}